// VectorQuantizer_55954833932991
// MI455X (gfx1250) — compile-verified
//
#include <hip/hip_runtime.h>

typedef __attribute__((ext_vector_type(2))) float v2f;
typedef __attribute__((ext_vector_type(8))) float v8f;

#define NUM_EMB 8192
#define DIM 64
#define ROWS 32768               // 32 * 1024
#define TILE_N 16
#define NUM_TILES (NUM_EMB / TILE_N)   // 512
#define ROWS_PER_BLOCK 128             // 8 waves * 16 rows
#define LDS_STRIDE 66                  // 64 + 2 pad: lanes hit distinct banks

// -------------------------------------------------------------------------
// Kernel 0: codebook row norms  ||e_k||^2  -> ws
// -------------------------------------------------------------------------
__global__ void vq_enorm_kernel(const float* __restrict__ cb,
                                float* __restrict__ enorm) {
  int r = blockIdx.x * blockDim.x + threadIdx.x;
  if (r >= NUM_EMB) return;
  const float4* p = (const float4*)(cb + r * DIM);
  float s = 0.f;
#pragma unroll
  for (int i = 0; i < DIM / 4; ++i) {
    float4 v = p[i];
    s += v.x * v.x + v.y * v.y + v.z * v.z + v.w * v.w;
  }
  enorm[r] = s;
}

// -------------------------------------------------------------------------
// Kernel 1: fused  (X . E^T)  +  argmin_k (||e_k||^2 - 2 x.e_k)
// 256 threads = 8 waves; each wave owns 16 rows of X as WMMA A-fragments.
// Codebook tiles (16 rows x 64 f32) double-buffered in LDS, shared by waves.
// -------------------------------------------------------------------------
__global__ __launch_bounds__(256) void vq_argmin_kernel(
    const float* __restrict__ x, const float* __restrict__ cb,
    const float* __restrict__ enorm, int* __restrict__ out_idx) {
  __shared__ float ldsE[2][TILE_N][LDS_STRIDE];

  const int tid   = threadIdx.x;
  const int lane  = tid & 31;
  const int wave  = tid >> 5;
  const int m     = lane & 15;            // A-frag row within slab / B column
  const int khalf = (lane >> 4) << 1;     // lanes 0-15: K+{0,1}; 16-31: K+{2,3}
  const int rowBase = blockIdx.x * ROWS_PER_BLOCK + wave * 16;

  // ---- load A fragments: 16 chunks of K=4 (V_WMMA_F32_16X16X4_F32 layout):
  //      lanes 0-15 hold (M=lane, K=4c+0/1), lanes 16-31 hold (M=lane-16, K=4c+2/3)
  v2f a[16];
#pragma unroll
  for (int c = 0; c < 16; ++c) {
    const float* p = x + (rowBase + m) * DIM + 4 * c + khalf;
    a[c] = *(const v2f*)p;
  }

  float minv[8];
  int   mini[8];
#pragma unroll
  for (int v = 0; v < 8; ++v) { minv[v] = 3.4e38f; mini[v] = 0; }

  // cooperative tile stage: 1024 floats / 256 threads = one float4 each
  const float4* gsrc = (const float4*)cb;
  auto load_tile = [&](int t, int buf) {
    float4 v = gsrc[t * (TILE_N * DIM / 4) + tid];
    int r = tid >> 4;                // codebook row within tile
    int c = (tid & 15) * 4;          // starting column
    ldsE[buf][r][c + 0] = v.x;
    ldsE[buf][r][c + 1] = v.y;
    ldsE[buf][r][c + 2] = v.z;
    ldsE[buf][r][c + 3] = v.w;
  };

  load_tile(0, 0);
  __syncthreads();

  for (int t = 0; t < NUM_TILES; ++t) {
    const int buf = t & 1;
    if (t + 1 < NUM_TILES) load_tile(t + 1, buf ^ 1);
    if (t + 2 < NUM_TILES)
      __builtin_prefetch(cb + (size_t)(t + 2) * TILE_N * DIM + tid * 4, 0, 0);

    // ---- 16 chained f32 WMMAs: acc(16x16) += A(16x4) * B(4x16)
    //      B layout: VGPR0 lanes 0-15 = (K=0,N=lane), lanes 16-31 = (K=2,N=lane-16)
    //                VGPR1 holds K=1 / K=3  -> b = E[n0+n, 4c+khalf .. +1]
    v8f acc = {};
#pragma unroll
    for (int c = 0; c < 16; ++c) {
      v2f b;
      b.x = ldsE[buf][m][4 * c + khalf + 0];
      b.y = ldsE[buf][m][4 * c + khalf + 1];
      acc = __builtin_amdgcn_wmma_f32_16x16x4_f32(
          /*neg_a=*/false, a[c], /*neg_b=*/false, b,
          /*c_mod=*/(short)0, acc, /*reuse_a=*/false, /*reuse_b=*/false);
    }

    // ---- epilogue: dist' = ||e||^2 - 2 * (x.e); running argmin per C slot
    const float en   = enorm[t * TILE_N + m];
    const int   cand = t * TILE_N + m;
#pragma unroll
    for (int v = 0; v < 8; ++v) {
      float d = fmaf(-2.0f, acc[v], en);
      if (d < minv[v]) { minv[v] = d; mini[v] = cand; }
    }
    __syncthreads();
  }

  // ---- reduce across the 16 columns held by each lane group (stays in-half)
#pragma unroll
  for (int mask = 8; mask >= 1; mask >>= 1) {
#pragma unroll
    for (int v = 0; v < 8; ++v) {
      float ov = __shfl_xor(minv[v], mask, 32);
      int   oi = __shfl_xor(mini[v], mask, 32);
      if (ov < minv[v] || (ov == minv[v] && oi < mini[v])) {
        minv[v] = ov; mini[v] = oi;
      }
    }
  }

  // C layout: lanes 0-15 hold rows M=v (0..7); lanes 16-31 hold rows M=8+v
  if ((lane & 15) == 0) {
    int rbase = rowBase + ((lane >> 4) << 3);
#pragma unroll
    for (int v = 0; v < 8; ++v) out_idx[rbase + v] = mini[v];
  }
}

// -------------------------------------------------------------------------
// Kernel 2: gather quantized rows, write indices (as f32), accumulate SSE
// 4 rows per 256-thread block (64 lanes per row).
// -------------------------------------------------------------------------
__global__ __launch_bounds__(256) void vq_gather_loss_kernel(
    const float* __restrict__ x, const float* __restrict__ cb,
    const int* __restrict__ idx, float* __restrict__ out_q,
    float* __restrict__ out_idx_f, float* __restrict__ acc) {
  int tid = threadIdx.x;
  int row = blockIdx.x * 4 + (tid >> 6);
  int col = tid & 63;
  int k = idx[row];
  float q  = cb[(size_t)k * DIM + col];
  float xi = x[(size_t)row * DIM + col];
  out_q[(size_t)row * DIM + col] = q;
  if (col == 0) out_idx_f[row] = (float)k;
  float d = q - xi;
  float s = d * d;
#pragma unroll
  for (int mask = 16; mask >= 1; mask >>= 1) s += __shfl_xor(s, mask, 32);
  if ((tid & 31) == 0) atomicAdd(acc, s);
}

// -------------------------------------------------------------------------
// Kernel 3: loss = (1 + 0.25) * SSE / numel   (both loss terms equal in value)
// -------------------------------------------------------------------------
__global__ void vq_finalize_kernel(const float* __restrict__ acc,
                                   float* __restrict__ out_loss) {
  *out_loss = 1.25f * (*acc) / (float)((size_t)ROWS * DIM);
}

// -------------------------------------------------------------------------
extern "C" void kernel_launch(void* const* d_in, const int* in_sizes, int n_in,
                              void* d_out, int out_size, void* d_ws,
                              size_t ws_size, hipStream_t stream) {
  const float* x  = (const float*)d_in[0];   // [32,1024,64] f32
  const float* cb = (const float*)d_in[1];   // [8192,64]    f32

  float* out      = (float*)d_out;
  float* q_out    = out;                         // 32768*64 floats
  float* loss_out = out + (size_t)ROWS * DIM;    // 1 float
  float* idx_out  = out + (size_t)ROWS * DIM + 1;// 32768 floats

  float* ws     = (float*)d_ws;
  float* enorm  = ws;                        // 8192 floats
  int*   idx_ws = (int*)(ws + NUM_EMB);      // 32768 ints
  float* acc    = ws + NUM_EMB + ROWS;       // 1 float

  hipMemsetAsync(acc, 0, sizeof(float), stream);
  vq_enorm_kernel<<<NUM_EMB / 256, 256, 0, stream>>>(cb, enorm);
  vq_argmin_kernel<<<ROWS / ROWS_PER_BLOCK, 256, 0, stream>>>(x, cb, enorm,
                                                              idx_ws);
  vq_gather_loss_kernel<<<ROWS / 4, 256, 0, stream>>>(x, cb, idx_ws, q_out,
                                                      idx_out, acc);
  vq_finalize_kernel<<<1, 1, 0, stream>>>(acc, loss_out);
}